// NeuDIFRefineSpatialGraph_45019847197209
// MI455X (gfx1250) — compile-verified
//
#include <hip/hip_runtime.h>
#include <hip/hip_bf16.h>
#include <math.h>

#define BB 64
#define WW 16
#define CC 64
#define TT 1024
#define DEM 64
#define WT (WW * TT) /* 16384 */

typedef __attribute__((ext_vector_type(2))) float v2f;
typedef __attribute__((ext_vector_type(8))) float v8f;

// ---------------------------------------------------------------------------
// Kernel 1: stat[bwc] = 0.5*(mean_T + energy_T) of x row. Memory-bound pass.
// ---------------------------------------------------------------------------
__global__ __launch_bounds__(256) void stat_kernel(const float* __restrict__ x,
                                                   float* __restrict__ stat) {
  const int row = blockIdx.x;  // (b*W + w)*C + c
  const float* __restrict__ p = x + (size_t)row * TT;
  float s = 0.f, q = 0.f;
  for (int t = threadIdx.x; t < TT; t += 256) {
    float v = p[t];
    s += v;
    q += v * v;
  }
  // wave32 reduction
  for (int off = 16; off > 0; off >>= 1) {
    s += __shfl_down(s, off, 32);
    q += __shfl_down(q, off, 32);
  }
  __shared__ float ls[8], lq[8];
  const int wid = threadIdx.x >> 5, lid = threadIdx.x & 31;
  if (lid == 0) { ls[wid] = s; lq[wid] = q; }
  __syncthreads();
  if (threadIdx.x == 0) {
    float S = 0.f, Q = 0.f;
    for (int i = 0; i < 8; ++i) { S += ls[i]; Q += lq[i]; }
    stat[row] = (S + Q) * (0.5f / (float)TT);
  }
}

// ---------------------------------------------------------------------------
// Kernel 2: Z[c,e] = mean over (b,w) of tanh(stat[b,w,c] * Wp[e])
// ---------------------------------------------------------------------------
__global__ __launch_bounds__(64) void zmean_kernel(const float* __restrict__ stat,
                                                   const float* __restrict__ Wp,
                                                   float* __restrict__ Z) {
  const int c = blockIdx.x;
  const int e = threadIdx.x;
  const float wpe = Wp[e];
  float a0 = 0.f, a1 = 0.f, a2 = 0.f, a3 = 0.f;
  for (int bw = 0; bw < BB * WW; bw += 4) {
    a0 += tanhf(stat[(bw + 0) * CC + c] * wpe);
    a1 += tanhf(stat[(bw + 1) * CC + c] * wpe);
    a2 += tanhf(stat[(bw + 2) * CC + c] * wpe);
    a3 += tanhf(stat[(bw + 3) * CC + c] * wpe);
  }
  Z[c * DEM + e] = (a0 + a1 + a2 + a3) * (1.0f / (float)(BB * WW));
}

// ---------------------------------------------------------------------------
// Kernel 3: build A (64x64) and A2 = A @ A. Single workgroup, everything in LDS.
// ---------------------------------------------------------------------------
__device__ __forceinline__ float gelu_exact(float v) {
  return 0.5f * v * (1.0f + erff(v * 0.70710678118654752f));
}
__device__ __forceinline__ float softplus_f(float v) {
  return (v > 20.0f) ? v : log1pf(expf(v));
}
__device__ __forceinline__ float gumbel_hash(unsigned idx) {
  // deterministic per-(i,j) Gumbel noise (stand-in for jax threefry key 42)
  unsigned z = idx * 0x9E3779B9u + 0x85EBCA6Bu;
  z ^= z >> 16; z *= 0x7FEB352Du;
  z ^= z >> 15; z *= 0x846CA68Bu;
  z ^= z >> 16;
  float u = ((float)(z >> 8) + 0.5f) * (1.0f / 16777216.0f);  // (0,1)
  return -logf(-logf(u));
}

__global__ __launch_bounds__(256) void build_graph_kernel(
    const float* __restrict__ Zg,
    const float* __restrict__ w1, const float* __restrict__ b1,
    const float* __restrict__ w2, const float* __restrict__ b2,
    const float* __restrict__ beta_p, const float* __restrict__ gamma_p,
    const float* __restrict__ temp_p,
    float* __restrict__ A2_out,    // workspace: A@A for fused propagation
    float* __restrict__ A_out) {   // d_out tail: final A
  __shared__ float Zs[CC * DEM];   // Z, later reused as generic scratch
  __shared__ float Ds[CC * CC];
  __shared__ float As[CC * CC];
  __shared__ float sq[CC];
  __shared__ float dinv[CC];
  __shared__ float w1s[16], b1s[16], w2s[16];
  __shared__ unsigned msk0[CC], msk1[CC];
  const int tid = threadIdx.x;

  if (tid < 16) { w1s[tid] = w1[tid]; b1s[tid] = b1[tid]; w2s[tid] = w2[tid]; }
  for (int i = tid; i < CC * DEM; i += 256) Zs[i] = Zg[i];
  __syncthreads();

  if (tid < CC) {
    float s = 0.f;
    for (int e = 0; e < DEM; ++e) { float v = Zs[tid * DEM + e]; s += v * v; }
    sq[tid] = s;
  }
  __syncthreads();

  // D2 = relu(sq_i + sq_j - 2 Z Z^T) * 0.5 ; A = I
  for (int idx = tid; idx < CC * CC; idx += 256) {
    const int i = idx >> 6, j = idx & 63;
    float dot = 0.f;
    for (int e = 0; e < DEM; ++e) dot += Zs[i * DEM + e] * Zs[j * DEM + e];
    Ds[idx] = 0.5f * fmaxf(sq[i] + sq[j] - 2.0f * dot, 0.0f);
    As[idx] = (i == j) ? 1.0f : 0.0f;
  }
  __syncthreads();

  const float beta = beta_p[0], gamma = gamma_p[0], b2v = b2[0];
  for (int step = 0; step < 2; ++step) {
    for (int idx = tid; idx < CC * CC; idx += 256) {
      float a = As[idx] - 0.2f * Ds[idx];
      float hsum = b2v;
#pragma unroll
      for (int k = 0; k < 16; ++k)
        hsum += w2s[k] * gelu_exact(a * w1s[k] + b1s[k]);
      a += beta * (softplus_f(hsum) * gamma);
      Zs[idx] = a;
    }
    __syncthreads();
    for (int idx = tid; idx < CC * CC; idx += 256) {
      const int i = idx >> 6, j = idx & 63;
      const float a = 0.5f * (Zs[i * 64 + j] + Zs[j * 64 + i]);
      As[idx] = fmaxf(a, 0.0f) + ((i == j) ? 1.0f : 0.0f);
    }
    __syncthreads();
  }

  // Gumbel-perturbed scores
  const float invT = 1.0f / fmaxf(temp_p[0], 1e-6f);
  for (int idx = tid; idx < CC * CC; idx += 256)
    Zs[idx] = (As[idx] + gumbel_hash((unsigned)idx)) * invT;
  __syncthreads();

  // top-16 per row via iterated argmax with 64-bit chosen mask
  if (tid < CC) {
    unsigned m0 = 0u, m1 = 0u;
    for (int pick = 0; pick < 16; ++pick) {
      float best = -3.402823466e+38f;
      int bj = 0;
      for (int j = 0; j < CC; ++j) {
        const unsigned taken =
            (j < 32) ? ((m0 >> j) & 1u) : ((m1 >> (j - 32)) & 1u);
        const float v = Zs[tid * 64 + j];
        if (!taken && v > best) { best = v; bj = j; }
      }
      if (bj < 32) m0 |= (1u << bj); else m1 |= (1u << (bj - 32));
    }
    msk0[tid] = m0; msk1[tid] = m1;
  }
  __syncthreads();

  for (int idx = tid; idx < CC * CC; idx += 256) {
    const int i = idx >> 6, j = idx & 63;
    const unsigned bit =
        (j < 32) ? ((msk0[i] >> j) & 1u) : ((msk1[i] >> (j - 32)) & 1u);
    Zs[idx] = bit ? As[idx] : 0.0f;
  }
  __syncthreads();
  for (int idx = tid; idx < CC * CC; idx += 256) {
    const int i = idx >> 6, j = idx & 63;
    const float a = 0.5f * (Zs[i * 64 + j] + Zs[j * 64 + i]);
    As[idx] = fmaxf(a, 0.0f) + ((i == j) ? 1.0f : 0.0f);
  }
  __syncthreads();

  if (tid < CC) {
    float d = 0.f;
    for (int j = 0; j < CC; ++j) d += As[tid * 64 + j];
    dinv[tid] = rsqrtf(fmaxf(d, 1e-6f));
  }
  __syncthreads();

  // final A = Dn^-1/2 A Dn^-1/2 + I ; emit A and keep in LDS for A2
  for (int idx = tid; idx < CC * CC; idx += 256) {
    const int i = idx >> 6, j = idx & 63;
    const float a = dinv[i] * As[idx] * dinv[j] + ((i == j) ? 1.0f : 0.0f);
    Zs[idx] = a;
    A_out[idx] = a;
  }
  __syncthreads();

  // A2 = A @ A   (folds the two propagation steps into one memory pass)
  for (int idx = tid; idx < CC * CC; idx += 256) {
    const int i = idx >> 6, j = idx & 63;
    float s = 0.f;
    for (int k = 0; k < CC; ++k) s += Zs[i * 64 + k] * Zs[k * 64 + j];
    A2_out[idx] = s;
  }
}

// ---------------------------------------------------------------------------
// Kernel 4: fused propagation out[b,:,wT..] = A2 (64x64) @ x[b,w] (64x1024)
// One wave per 64x16 output slab; f32 WMMA 16x16x4, K=64 in 16 steps.
// ---------------------------------------------------------------------------
__global__ __launch_bounds__(32) void prop_kernel(const float* __restrict__ x,
                                                  const float* __restrict__ A2,
                                                  float* __restrict__ out) {
  const int nb = blockIdx.x;   // 0..63 column tile (16 cols)
  const int bw = blockIdx.y;   // 0..1023  (b*W + w)
  const int b = bw >> 4;
  const int w = bw & 15;
  const int lane = (int)threadIdx.x;
  const int half = lane >> 4;        // 0: K pair {0,1}, 1: K pair {2,3}
  const int l16 = lane & 15;
  const int koff = half << 1;
  const int ncol = nb * 16 + l16;
  const float* __restrict__ X = x + (size_t)bw * (CC * TT);

  v8f acc0 = {}, acc1 = {}, acc2 = {}, acc3 = {};
#pragma unroll
  for (int s = 0; s < 16; ++s) {
    const int kb = (s << 2) + koff;  // even -> 8B-aligned float2 loads of A2
    v2f bf;
    bf.x = X[(size_t)kb * TT + ncol];
    bf.y = X[(size_t)(kb + 1) * TT + ncol];
    const v2f a0 = *reinterpret_cast<const v2f*>(A2 + (l16 + 0) * CC + kb);
    const v2f a1 = *reinterpret_cast<const v2f*>(A2 + (l16 + 16) * CC + kb);
    const v2f a2 = *reinterpret_cast<const v2f*>(A2 + (l16 + 32) * CC + kb);
    const v2f a3 = *reinterpret_cast<const v2f*>(A2 + (l16 + 48) * CC + kb);
    acc0 = __builtin_amdgcn_wmma_f32_16x16x4_f32(false, a0, false, bf, (short)0,
                                                 acc0, false, false);
    acc1 = __builtin_amdgcn_wmma_f32_16x16x4_f32(false, a1, false, bf, (short)0,
                                                 acc1, false, false);
    acc2 = __builtin_amdgcn_wmma_f32_16x16x4_f32(false, a2, false, bf, (short)0,
                                                 acc2, false, false);
    acc3 = __builtin_amdgcn_wmma_f32_16x16x4_f32(false, a3, false, bf, (short)0,
                                                 acc3, false, false);
  }

  // D layout: VGPR r -> M = r (lanes 0-15) / M = 8+r (lanes 16-31), N = lane&15
  float* __restrict__ o =
      out + (size_t)b * (CC * WT) + (size_t)w * TT + (size_t)(nb * 16 + l16);
#pragma unroll
  for (int r = 0; r < 8; ++r) {
    const int m = r + half * 8;
    o[(size_t)(m + 0)  * WT] = acc0[r];
    o[(size_t)(m + 16) * WT] = acc1[r];
    o[(size_t)(m + 32) * WT] = acc2[r];
    o[(size_t)(m + 48) * WT] = acc3[r];
  }
}

// ---------------------------------------------------------------------------
extern "C" void kernel_launch(void* const* d_in, const int* in_sizes, int n_in,
                              void* d_out, int out_size, void* d_ws,
                              size_t ws_size, hipStream_t stream) {
  const float* x    = (const float*)d_in[0];
  const float* Wp   = (const float*)d_in[1];
  const float* w1   = (const float*)d_in[2];
  const float* b1   = (const float*)d_in[3];
  const float* w2   = (const float*)d_in[4];
  const float* b2   = (const float*)d_in[5];
  const float* beta = (const float*)d_in[6];
  const float* gam  = (const float*)d_in[7];
  const float* temp = (const float*)d_in[8];
  float* out = (float*)d_out;

  // workspace layout (floats): stat[65536] | Z[4096] | A2[4096]
  float* stat = (float*)d_ws;
  float* Zg   = stat + BB * WW * CC;
  float* A2   = Zg + CC * DEM;

  stat_kernel<<<BB * WW * CC, 256, 0, stream>>>(x, stat);
  zmean_kernel<<<CC, 64, 0, stream>>>(stat, Wp, Zg);
  build_graph_kernel<<<1, 256, 0, stream>>>(
      Zg, w1, b1, w2, b2, beta, gam, temp, A2,
      out + (size_t)BB * CC * WT);  // A goes after xr in d_out
  prop_kernel<<<dim3(TT / 16, BB * WW), 32, 0, stream>>>(x, A2, out);
}